// GNN_39238821216991
// MI455X (gfx1250) — compile-verified
//
#include <hip/hip_runtime.h>
#include <hip/hip_bf16.h>

typedef __attribute__((ext_vector_type(2))) float v2f;
typedef __attribute__((ext_vector_type(8))) float v8f;

#define DIM 64
#define NEG_SLOPE 0.01f

// ---------------------------------------------------------------------------
// fill: grid-stride constant fill
// ---------------------------------------------------------------------------
__global__ void gnn_fill(float* __restrict__ p, long n, float v) {
    long i = (long)blockIdx.x * blockDim.x + threadIdx.x;
    long stride = (long)gridDim.x * blockDim.x;
    for (; i < n; i += stride) p[i] = v;
}

// ---------------------------------------------------------------------------
// degree: deg[dst] += 1 per edge (deg pre-filled with 1.0 for self-loop)
// ---------------------------------------------------------------------------
__global__ void gnn_degree(const int* __restrict__ ei, float* __restrict__ deg,
                           int nEdges) {
    int e = blockIdx.x * blockDim.x + threadIdx.x;
    if (e < nEdges) atomicAdd(&deg[ei[nEdges + e]], 1.0f);
}

// dinv = rsqrt(deg) in place (deg >= 1 always because of self-loops)
__global__ void gnn_rsqrt(float* __restrict__ deg, int n) {
    int i = blockIdx.x * blockDim.x + threadIdx.x;
    if (i < n) deg[i] = rsqrtf(deg[i]);
}

// ---------------------------------------------------------------------------
// H[M,64] = A[M,64] @ W[64,64] using V_WMMA_F32_16X16X4_F32 (full fp32).
// One wave per 16x16 output tile; 16 WMMA steps cover K=64.
// A-frag (16x4 f32): lanes 0-15 hold M=lane, K={k0,k0+1}; lanes 16-31 K={k0+2,k0+3}
// B-frag (4x16 f32): lanes hold N=lane&15; VGPR0 K=k0(+2 hi half), VGPR1 K=k0+1(+2)
// C/D (16x16 f32): VGPR r -> M = r + 8*(lane>>4), N = lane&15
// Out-of-range rows are CLAMPED (not predicated) so the inner loop has no
// divergence: EXEC stays all-1s, loads are branch-free b64/b32 clauses.
// ---------------------------------------------------------------------------
__global__ void gnn_gemm64_wmma(const float* __restrict__ A,
                                const float* __restrict__ W,
                                float* __restrict__ H, int M) {
    const int lane = threadIdx.x & 31;
    const int waveId = (blockIdx.x * blockDim.x + threadIdx.x) >> 5;
    const int mTiles = (M + 15) >> 4;
    if (waveId >= mTiles * 4) return;           // wave-uniform: EXEC stays all-1s
    const int mt = waveId >> 2;
    const int nt = waveId & 3;

    const int half = lane >> 4;                 // K half-split for A/B frags
    const int l = lane & 15;

    int m = mt * 16 + l;
    if (m >= M) m = M - 1;                      // clamp: in-bounds, branch-free
    const float2* __restrict__ aPair =
        (const float2*)(A + (size_t)m * DIM) + half;   // pair index = k0/2 + half
    const float* __restrict__ wCol = W + nt * 16 + l;

    v8f c = {};
#pragma unroll
    for (int k0 = 0; k0 < DIM; k0 += 4) {
        const float2 ap = aPair[k0 >> 1];       // one global_load_b64
        v2f a; a.x = ap.x; a.y = ap.y;
        const int ka = k0 + 2 * half;
        v2f b;
        b.x = wCol[(size_t)(ka + 0) * DIM];
        b.y = wCol[(size_t)(ka + 1) * DIM];
        c = __builtin_amdgcn_wmma_f32_16x16x4_f32(
                /*neg_a=*/false, a, /*neg_b=*/false, b,
                /*c_mod=*/(short)0, c, /*reuse_a=*/false, /*reuse_b=*/false);
    }

#pragma unroll
    for (int r = 0; r < 8; ++r) {
        const int row = mt * 16 + r + 8 * half;
        if (row < M) H[(size_t)row * DIM + nt * 16 + l] = c[r];
    }
}

// ---------------------------------------------------------------------------
// Edge scatter: one wave32 per edge; lane owns a float2 of the 64-d message.
// agg[dst] += h[src] * dinv[src]*dinv[dst]
// ---------------------------------------------------------------------------
__global__ void gnn_scatter(const int* __restrict__ ei,
                            const float* __restrict__ h,
                            const float* __restrict__ dinv,
                            float* __restrict__ agg, int nEdges) {
    const int w = (blockIdx.x * blockDim.x + threadIdx.x) >> 5;
    const int lane = threadIdx.x & 31;
    if (w >= nEdges) return;                    // wave-uniform
    const int s = ei[w];
    const int d = ei[nEdges + w];
    const float nrm = dinv[s] * dinv[d];
    const float2 hv = *(const float2*)(h + (size_t)s * DIM + lane * 2);
    float* out = agg + (size_t)d * DIM + lane * 2;
    atomicAdd(out + 0, hv.x * nrm);
    atomicAdd(out + 1, hv.y * nrm);
}

// ---------------------------------------------------------------------------
// Layer-1 finalize: agg = leaky_relu(agg + h*dinv^2 + bias)   (in place)
// ---------------------------------------------------------------------------
__global__ void gnn_finalize1(float* __restrict__ agg,
                              const float* __restrict__ h,
                              const float* __restrict__ dinv,
                              const float* __restrict__ bias, long n) {
    long i = (long)blockIdx.x * blockDim.x + threadIdx.x;
    if (i >= n) return;
    const int node = (int)(i >> 6);
    const int ch = (int)(i & 63);
    const float di = dinv[node];
    const float v = agg[i] + h[i] * di * di + bias[ch];
    agg[i] = (v >= 0.0f) ? v : NEG_SLOPE * v;
}

// ---------------------------------------------------------------------------
// Layer-2 finalize fused with mean-pool sum accumulation.
// pool[batch[node]][ch] += agg + h*dinv^2 + bias
// ---------------------------------------------------------------------------
__global__ void gnn_finalize2_pool(const float* __restrict__ agg,
                                   const float* __restrict__ h,
                                   const float* __restrict__ dinv,
                                   const float* __restrict__ bias,
                                   const int* __restrict__ batch,
                                   float* __restrict__ pool, long n) {
    long i = (long)blockIdx.x * blockDim.x + threadIdx.x;
    if (i >= n) return;
    const int node = (int)(i >> 6);
    const int ch = (int)(i & 63);
    const float di = dinv[node];
    const float v = agg[i] + h[i] * di * di + bias[ch];
    atomicAdd(&pool[(size_t)batch[node] * DIM + ch], v);
}

__global__ void gnn_count(const int* __restrict__ batch, float* __restrict__ cnt,
                          int n) {
    int i = blockIdx.x * blockDim.x + threadIdx.x;
    if (i < n) atomicAdd(&cnt[batch[i]], 1.0f);
}

// ---------------------------------------------------------------------------
// Head: out[g][j] = (sum_d pool[g][d] * fc_w[d][j]) / max(cnt[g],1) + fc_b[j]
// ---------------------------------------------------------------------------
__global__ void gnn_head(const float* __restrict__ pool,
                         const float* __restrict__ cnt,
                         const float* __restrict__ fcw,
                         const float* __restrict__ fcb,
                         float* __restrict__ out, int nGraphs) {
    const int i = blockIdx.x * blockDim.x + threadIdx.x;
    if (i >= nGraphs * 3) return;
    const int g = i / 3;
    const int j = i % 3;
    float acc = 0.0f;
#pragma unroll
    for (int d = 0; d < DIM; ++d) acc += pool[(size_t)g * DIM + d] * fcw[d * 3 + j];
    float c = cnt[g];
    c = (c > 1.0f) ? c : 1.0f;
    out[i] = acc / c + fcb[j];
}

// ---------------------------------------------------------------------------
extern "C" void kernel_launch(void* const* d_in, const int* in_sizes, int n_in,
                              void* d_out, int out_size, void* d_ws, size_t ws_size,
                              hipStream_t stream) {
    const float* x    = (const float*)d_in[0];
    const int*   ei   = (const int*)  d_in[1];
    const int*   bat  = (const int*)  d_in[2];
    const float* W1   = (const float*)d_in[3];
    const float* b1   = (const float*)d_in[4];
    const float* W2   = (const float*)d_in[5];
    const float* b2   = (const float*)d_in[6];
    const float* fcw  = (const float*)d_in[7];
    const float* fcb  = (const float*)d_in[8];
    float* out = (float*)d_out;

    const int N  = in_sizes[0] / DIM;       // 100000
    const int E  = in_sizes[1] / 2;         // 1200000
    const int G  = out_size / 3;            // 128
    const long NF = (long)N * DIM;          // 6.4M

    // workspace layout (floats)
    float* ws = (float*)d_ws;
    const long nAlign   = ((long)N + 255) & ~255L;
    float* dinv = ws;                       // N
    float* h    = dinv + nAlign;            // N*64
    float* agg1 = h + NF;                   // N*64
    float* agg2 = agg1 + NF;                // N*64
    float* pool = agg2 + NF;                // G*64
    float* cnt  = pool + (long)G * DIM;     // G

    const int T = 256;
    // ---- init ----
    gnn_fill<<<1024, T, 0, stream>>>(dinv, N, 1.0f);       // deg starts at 1 (self-loop)
    gnn_fill<<<2048, T, 0, stream>>>(agg1, NF, 0.0f);
    gnn_fill<<<2048, T, 0, stream>>>(agg2, NF, 0.0f);
    gnn_fill<<<32,   T, 0, stream>>>(pool, (long)G * DIM, 0.0f);
    gnn_fill<<<1,    T, 0, stream>>>(cnt, G, 0.0f);

    // ---- degrees / symmetric norm ----
    gnn_degree<<<(E + T - 1) / T, T, 0, stream>>>(ei, dinv, E);
    gnn_rsqrt <<<(N + T - 1) / T, T, 0, stream>>>(dinv, N);

    const int mTiles = (N + 15) / 16;
    const int gemmBlocks = (mTiles * 4 + (T / 32) - 1) / (T / 32);
    const int scatBlocks = (int)(((long)E * 32 + T - 1) / T);

    // ---- layer 1 ----
    gnn_gemm64_wmma<<<gemmBlocks, T, 0, stream>>>(x, W1, h, N);
    gnn_scatter    <<<scatBlocks, T, 0, stream>>>(ei, h, dinv, agg1, E);
    gnn_finalize1  <<<(int)((NF + T - 1) / T), T, 0, stream>>>(agg1, h, dinv, b1, NF);

    // ---- layer 2 ----
    gnn_gemm64_wmma<<<gemmBlocks, T, 0, stream>>>(agg1, W2, h, N);
    gnn_scatter    <<<scatBlocks, T, 0, stream>>>(ei, h, dinv, agg2, E);
    gnn_count      <<<(N + T - 1) / T, T, 0, stream>>>(bat, cnt, N);
    gnn_finalize2_pool<<<(int)((NF + T - 1) / T), T, 0, stream>>>(agg2, h, dinv, b2,
                                                                  bat, pool, NF);

    // ---- head ----
    gnn_head<<<(G * 3 + T - 1) / T, T, 0, stream>>>(pool, cnt, fcw, fcb, out, G);
}